// ModelNew_3556232921876
// MI455X (gfx1250) — compile-verified
//
#include <hip/hip_runtime.h>
#include <hip/hip_bf16.h>

typedef __attribute__((ext_vector_type(2))) float v2f;
typedef __attribute__((ext_vector_type(8))) float v8f;
typedef __attribute__((ext_vector_type(2))) int   v2i;
typedef __attribute__((ext_vector_type(4))) unsigned int v4u;
typedef __attribute__((ext_vector_type(8))) int   v8i;
typedef __attribute__((ext_vector_type(4))) int   v4i;

#define C_IN 8
#define C_OUT 32
#define IN_W 48
#define OUT_W 46
#define KTOT 216              // 8 * 27
#define WSTRIDE 40            // 2*40 = 16 mod 64 -> half-waves hit disjoint bank groups
#define XSTRIDE 48            // TDM writes the tile linearly (row = 48 dwords)
#define OHB 23                // 46 / 2 oh-pairs

#if defined(__gfx1250__) && __has_builtin(__builtin_amdgcn_tensor_load_to_lds)
#define USE_TDM 1
#else
#define USE_TDM 0
#endif

// ---------------------------------------------------------------------------
// Zero the workspace accumulators (S1,S2 per (b,c)): 16*32*2 = 1024 floats
// ---------------------------------------------------------------------------
__global__ __launch_bounds__(256) void zero_ws_kernel(float* __restrict__ ws) {
    int i = blockIdx.x * blockDim.x + threadIdx.x;
    if (i < 16 * 32 * 2) ws[i] = 0.0f;
}

// ---------------------------------------------------------------------------
// Implicit-GEMM conv3d via V_WMMA_F32_16X16X4_F32, fused bias + HardSwish +
// per-channel sum / sum-of-squares accumulation.
// Block = 64 threads (2 waves, one per 16-channel M-tile).
// blockIdx.x encodes (b, od, oh_pair); each wave keeps 6 accumulators
// (2 oh x 3 ow-tiles) live so one A-fragment feeds 6 WMMAs.
// Input slab is staged LDS-side by the Tensor Data Mover (one 4D-tile DMA).
// ---------------------------------------------------------------------------
__global__ __launch_bounds__(64) void conv_sums_kernel(
    const float* __restrict__ x,     // [16, 8, 48, 48, 48]
    const float* __restrict__ w,     // [32, 8, 3, 3, 3]  (O-major, 216 per O)
    const float* __restrict__ bias,  // [32]
    float* __restrict__ partial)     // [16][32][2]  (S1, S2)
{
    __shared__ float Wlds[KTOT * WSTRIDE];              // W[k][c]   (~34.6 KB)
    __shared__ float Xlds[C_IN * 3 * 4 * XSTRIDE + 2];  // x slab + overreach slack
    __shared__ v2i   BoffT2[108];                       // per-half-wave K->offset pairs
    __shared__ float Sacc[C_OUT * 2];                   // per-channel S1, S2
    __shared__ float Bias_s[C_OUT];

    int* BoffT = (int*)BoffT2;

    const int tid = threadIdx.x;
    const int blk = blockIdx.x;
    const int ohb = blk % OHB;
    const int od  = (blk / OHB) % OUT_W;
    const int b   = blk / (OHB * OUT_W);
    const int oh0 = ohb * 2;

    const float* xb = x + (size_t)b * (C_IN * IN_W * IN_W * IN_W);

#if USE_TDM
    // ---- async tensor DMA: 4D tile (w=48, hr=4, kd=3, ci=8) -> LDS ----
    if (tid < 32) {   // wave 0 issues; TDM ignores EXEC, one op per wave
        unsigned long long ga =
            (unsigned long long)(size_t)(xb + ((size_t)od * IN_W + oh0) * IN_W);
        unsigned lv = (unsigned)(size_t)(void*)&Xlds[0];   // LDS byte address
        v4u g0 = { 1u,                                 // count=1
                   lv,                                 // lds_addr
                   (unsigned)ga,                       // global_addr[31:0]
                   (unsigned)((ga >> 32) & 0x01FFFFFFu) | (2u << 30) }; // type=2
        v8i g1 = { (int)(2u << 16),                    // data_size = 4B
                   (int)(IN_W << 16),                  // tensor_dim0 = 48
                   (int)(IN_W << 16),                  // tensor_dim1 = 48
                   (int)(IN_W << 16),                  // tile_dim0 = 48
                   (int)(4u | (3u << 16)),             // tile_dim1 = 4, tile_dim2 = 3
                   (int)IN_W,                          // dim0_stride = 48
                   (int)((unsigned)(IN_W * IN_W) << 16),   // dim1_stride = 2304
                   0 };
        v4i g2 = { IN_W,                               // tensor_dim2 = 48
                   C_IN,                               // tensor_dim3 = 8
                   IN_W * IN_W * IN_W,                 // dim2_stride = 110592
                   (int)(8u << 16) };                  // tile_dim3 = 8
        v4i g3 = { 0, 0, 0, 0 };
        v8i g4 = { 0, 0, 0, 0, 0, 0, 0, 0 };           // reserved/extended words
        __builtin_amdgcn_tensor_load_to_lds(g0, g1, g2, g3, g4, 0);
    }
#endif

    // ---- stage weights transposed: Wlds[k*WSTRIDE + c] = w[c*216 + k] ----
    for (int i = tid; i < KTOT * C_OUT; i += 64) {
        int k = i >> 5, c = i & 31;
        Wlds[k * WSTRIDE + c] = w[c * KTOT + k];
    }
#if !USE_TDM
    // ---- fallback: manual staging of the x slab ----
    for (int i = tid; i < C_IN * 3 * 4 * IN_W; i += 64) {
        int wc = i % IN_W;
        int t  = i / IN_W;        // 0..95 -> (ci, kd, hr)
        int hr = t & 3;
        int t2 = t >> 2;          // 0..23
        int kd = t2 % 3;
        int ci = t2 / 3;
        Xlds[t * XSTRIDE + wc] =
            xb[((ci * IN_W + od + kd) * IN_W + (oh0 + hr)) * IN_W + wc];
    }
#endif
    // ---- build K -> Xlds word-offset table, grouped per half-wave so each
    //      step fetches its (k0, k0+1) pair with one ds_load_b64 ----
    for (int k = tid; k < KTOT; k += 64) {
        int ci = k / 27, r = k - ci * 27;
        int kd = r / 9,  q = r - kd * 9;
        int kh = q / 3,  kw = q - kh * 3;
        int val = ((ci * 3 + kd) * 4 + kh) * XSTRIDE + kw;
        int idx = ((k >> 1) & 1) * 108 + ((k >> 2) << 1) + (k & 1);
        BoffT[idx] = val;
    }
    Sacc[tid] = 0.0f;                       // 64 threads cover 64 entries
    if (tid < C_OUT) Bias_s[tid] = bias[tid];
#if USE_TDM
    if (tid < 32) __builtin_amdgcn_s_wait_tensorcnt(0);
#endif
    __syncthreads();

    const int lane = tid & 31;
    const int wave = tid >> 5;
    const int c0   = wave * 16;             // M-tile: channels c0..c0+15
    const int m    = lane & 15;             // row (A) / column (B,C) index
    const int hi   = lane >> 4;             // half-wave: owns K sub-pairs {0,1}/{2,3}

    // Per-lane LDS bases (word units)
    const int abase = (hi * 2) * WSTRIDE + c0 + m;   // + s*4*WSTRIDE per step
    const int tbase = hi * 108;                      // offset-pair table base
    int bbase[6];
    #pragma unroll
    for (int t = 0; t < 6; ++t)
        bbase[t] = (t / 3) * XSTRIDE + (t % 3) * 16 + m;   // hsel*XSTRIDE + ow0 + m

    v8f acc[6];
    #pragma unroll
    for (int t = 0; t < 6; ++t) acc[t] = (v8f){0,0,0,0,0,0,0,0};

    #pragma unroll 6
    for (int s = 0; s < 54; ++s) {                  // K = 4*s
        const v2i off = BoffT2[tbase / 2 + s];      // {boff(k0), boff(k0+1)}
        v2f a;
        a.x = Wlds[abase + s * (4 * WSTRIDE)];
        a.y = Wlds[abase + s * (4 * WSTRIDE) + WSTRIDE];
        #pragma unroll
        for (int t = 0; t < 6; ++t) {
            v2f bf;
            bf.x = Xlds[bbase[t] + off.x];
            bf.y = Xlds[bbase[t] + off.y];
            acc[t] = __builtin_amdgcn_wmma_f32_16x16x4_f32(
                false, a, false, bf, (short)0, acc[t], false, false);
        }
    }

    // ---- epilogue: bias + HardSwish, accumulate per-lane over all tiles ----
    float br[8];
    const int chb = c0 + (hi << 3);
    #pragma unroll
    for (int r = 0; r < 8; ++r) br[r] = Bias_s[chb + r];

    float sum1[8], sum2[8];
    #pragma unroll
    for (int r = 0; r < 8; ++r) { sum1[r] = 0.0f; sum2[r] = 0.0f; }

    #pragma unroll
    for (int t = 0; t < 6; ++t) {
        const bool colok = (t % 3 != 2) || (m < 14);   // last ow-tile: 14 cols
        #pragma unroll
        for (int r = 0; r < 8; ++r) {
            float yv = acc[t][r] + br[r];
            float t6 = fminf(fmaxf(yv + 3.0f, 0.0f), 6.0f);
            float u  = yv * t6;                        // 6 * hardswish
            sum1[r] += colok ? u : 0.0f;
            sum2[r] += colok ? u * u : 0.0f;
        }
    }
    // one LDS float-atomic flush per wave (16-way lane merge in hardware)
    #pragma unroll
    for (int r = 0; r < 8; ++r) {
        atomicAdd(&Sacc[(chb + r) * 2 + 0], sum1[r] * (1.0f / 6.0f));
        atomicAdd(&Sacc[(chb + r) * 2 + 1], sum2[r] * (1.0f / 36.0f));
    }
    __syncthreads();

    // ---- one global atomic per (channel, stat) per workgroup ----
    {
        int ch = tid >> 1, which = tid & 1;
        atomicAdd(&partial[(b * C_OUT + ch) * 2 + which], Sacc[ch * 2 + which]);
    }
}

// ---------------------------------------------------------------------------
// Finalize: GroupNorm statistics from S1/S2 + pooled output. 512 threads.
// ---------------------------------------------------------------------------
__global__ __launch_bounds__(512) void finalize_kernel(
    const float* __restrict__ partial,   // [16][32][2]
    const float* __restrict__ gnw,
    const float* __restrict__ gnb,
    float* __restrict__ out)             // [16*32]
{
    __shared__ float g1[64], g2[64];     // per (b, group)
    const int tid = threadIdx.x;         // 0..511 -> (b, c)
    if (tid < 64) { g1[tid] = 0.0f; g2[tid] = 0.0f; }
    __syncthreads();

    const int b = tid >> 5, c = tid & 31;
    const float s1 = partial[tid * 2 + 0];
    const float s2 = partial[tid * 2 + 1];
    const int g = b * 4 + (c >> 3);
    atomicAdd(&g1[g], s1);
    atomicAdd(&g2[g], s2);
    __syncthreads();

    const float S      = 46.0f * 46.0f * 46.0f;   // 97336
    const float invCnt = 1.0f / (8.0f * S);       // 8 channels per group
    const float mean   = g1[g] * invCnt;
    const float var    = g2[g] * invCnt - mean * mean;
    const float rs     = rsqrtf(var + 1e-5f);
    out[tid] = (s1 / S - mean) * rs * gnw[c] + gnb[c];
}

// ---------------------------------------------------------------------------
extern "C" void kernel_launch(void* const* d_in, const int* in_sizes, int n_in,
                              void* d_out, int out_size, void* d_ws, size_t ws_size,
                              hipStream_t stream) {
    const float* x    = (const float*)d_in[0];
    const float* w    = (const float*)d_in[1];
    const float* bias = (const float*)d_in[2];
    const float* gnw  = (const float*)d_in[3];
    const float* gnb  = (const float*)d_in[4];
    float* out     = (float*)d_out;
    float* partial = (float*)d_ws;     // 1024 floats

    zero_ws_kernel<<<4, 256, 0, stream>>>(partial);
    conv_sums_kernel<<<16 * OUT_W * OHB, 64, 0, stream>>>(x, w, bias, partial);
    finalize_kernel<<<1, 512, 0, stream>>>(partial, gnw, gnb, out);
}